// NTfm3D_9783935500930
// MI455X (gfx1250) — compile-verified
//
#include <hip/hip_runtime.h>

typedef __attribute__((ext_vector_type(2))) float v2f;
typedef __attribute__((ext_vector_type(8))) float v8f;

#define NB   16
#define NK   8
#define NHW  (480 * 640)          // 307200 pixels per batch
#define ITERS 8                   // 32-pixel groups per wave
#define WAVES_PER_BLOCK 8         // 256 threads, wave32
#define PIX_PER_WAVE (32 * ITERS) // 256
#define WAVES_PER_BATCH (NHW / PIX_PER_WAVE)                  // 1200
#define BLOCKS_PER_BATCH (WAVES_PER_BATCH / WAVES_PER_BLOCK)  // 150

// out[b,i,hw] = sum_k mask[b,k,hw] * ( sum_j R[b,k,i,j]*p[b,j,hw] + t[b,k,i] )
// GEMM form: out = A(16x32, rows>=3 zero) @ X(32xN), X[4k+j][n] = mask_k[n]*paug_j[n],
// chained V_WMMA_F32_16X16X4_F32 (one K=4 chunk per mask k, two 16-pixel tiles/wave).
// B operands are assembled by lane-swizzled global loads (no input-side permutes).
__global__ __launch_bounds__(256) void ntfm3d_wmma_kernel(
    const float* __restrict__ points,
    const float* __restrict__ masks,
    const float* __restrict__ transforms,
    float* __restrict__ out)
{
    const int lane = threadIdx.x & 31;
    const int wave = threadIdx.x >> 5;
    const int b = blockIdx.x / BLOCKS_PER_BATCH;
    const int waveInBatch = (blockIdx.x % BLOCKS_PER_BATCH) * WAVES_PER_BLOCK + wave;
    const int pixBase0 = waveInBatch * PIX_PER_WAVE;

    const float* __restrict__ P = points     + (size_t)b * 3  * NHW;
    const float* __restrict__ M = masks      + (size_t)b * NK * NHW;
    const float* __restrict__ T = transforms + (size_t)b * NK * 12;
    float*       __restrict__ O = out        + (size_t)b * 3  * NHW;

    const int  n   = lane & 15;        // B column / A row within the 16-wide tile
    const bool lo  = lane < 16;        // K-half select (kk 0/1 vs kk 2/3)
    const int  kk2 = (lane >> 4) * 2;

    // Per-batch A tiles: A[c] vgpr0 = T[.,c,m,kk2], vgpr1 = T[.,c,m,kk2+1]; rows m>=3 zero.
    v2f A[NK];
#pragma unroll
    for (int c = 0; c < NK; ++c) {
        float a0 = 0.0f, a1 = 0.0f;
        if (n < 3) {
            a0 = T[(c * 3 + n) * 4 + kk2 + 0];
            a1 = T[(c * 3 + n) * 4 + kk2 + 1];
        }
        A[c].x = a0;
        A[c].y = a1;
    }

    // Lane-swizzled constant offsets (in floats) relative to the pixel base:
    //   B-vgpr0 of a tile needs p0 (lanes 0-15, kk=0) / p2 (lanes 16-31, kk=2);
    //   B-vgpr1 needs p1 (lanes 0-15, kk=1) / constant 1.0 (lanes 16-31, kk=3).
    const int offA = (lo ? 0 : 2) * NHW + n;  // tile0; tile1 = +16
    const int offB = 1 * NHW + n;             // p1 (hi lanes load a duplicate, value unused)

    for (int it = 0; it < ITERS; ++it) {
        const int base = pixBase0 + it * 32;
        const float* __restrict__ Pit = P + base;

        float a0v = __builtin_nontemporal_load(Pit + offA);        // tile0 cols base..base+15
        float a1v = __builtin_nontemporal_load(Pit + offA + 16);   // tile1 cols base+16..+31
        float t0  = __builtin_nontemporal_load(Pit + offB);
        float t1  = __builtin_nontemporal_load(Pit + offB + 16);
        float b0v = lo ? t0 : 1.0f;
        float b1v = lo ? t1 : 1.0f;

        float mt0[NK], mt1[NK];
#pragma unroll
        for (int c = 0; c < NK; ++c) {
            const float* __restrict__ Mc = M + (size_t)c * NHW + base;
            mt0[c] = __builtin_nontemporal_load(Mc + n);        // both halves: same 16 values
            mt1[c] = __builtin_nontemporal_load(Mc + n + 16);
        }

        v8f D0 = {};
        v8f D1 = {};
#pragma unroll
        for (int c = 0; c < NK; ++c) {
            v2f B0; B0.x = mt0[c] * a0v; B0.y = mt0[c] * b0v;
            v2f B1; B1.x = mt1[c] * a1v; B1.y = mt1[c] * b1v;
            D0 = __builtin_amdgcn_wmma_f32_16x16x4_f32(
                     false, A[c], false, B0, (short)0, D0, false, false);
            D1 = __builtin_amdgcn_wmma_f32_16x16x4_f32(
                     false, A[c], false, B1, (short)0, D1, false, false);
        }

        // D rows 0..2 sit in accumulator vgprs 0..2, lanes 0..15. Merge tile1 into
        // the upper lane half for one full-width coalesced store per channel.
#pragma unroll
        for (int i = 0; i < 3; ++i) {
            float d1 = __shfl_xor(D1[i], 16);
            float v  = lo ? D0[i] : d1;
            __builtin_nontemporal_store(v, O + (size_t)i * NHW + base + lane);
        }
    }
}

extern "C" void kernel_launch(void* const* d_in, const int* in_sizes, int n_in,
                              void* d_out, int out_size, void* d_ws, size_t ws_size,
                              hipStream_t stream) {
    const float* points     = (const float*)d_in[0];
    const float* masks      = (const float*)d_in[1];
    const float* transforms = (const float*)d_in[2];
    float* out = (float*)d_out;

    dim3 grid(NB * BLOCKS_PER_BATCH);  // 2400 blocks
    dim3 block(256);                   // 8 wave32 waves
    ntfm3d_wmma_kernel<<<grid, block, 0, stream>>>(points, masks, transforms, out);
}